// PPLoss_3410204033167
// MI455X (gfx1250) — compile-verified
//
#include <hip/hip_runtime.h>
#include <hip/hip_bf16.h>
#include <math.h>

// Problem constants (match reference)
#define B_  128
#define C_  512
#define S_  8
#define L_  32
#define D_  128

#define EPS_NORM 1e-12f
#define EPS_PD   1e-6f
#define MARGIN   8.0f

typedef __attribute__((ext_vector_type(2))) float v2f;
typedef __attribute__((ext_vector_type(8))) float v8f;

#if defined(__gfx1250__) && __has_builtin(__builtin_amdgcn_wmma_f32_16x16x4_f32)
#define USE_WMMA 1
#else
#define USE_WMMA 0
#endif

// ---------------------------------------------------------------------------
// Kernel A: per-row inverse L2 norm. One wave32 per row (D=128 -> float4/lane).
// ---------------------------------------------------------------------------
__global__ __launch_bounds__(256)
void pp_rownorm_kernel(const float* __restrict__ feats, float* __restrict__ invn,
                       int nrows) {
    const int lane = threadIdx.x & 31;
    const int wave = threadIdx.x >> 5;
    const int row  = blockIdx.x * 8 + wave;
    if (row >= nrows) return;
    const float4 v = ((const float4*)(feats + (size_t)row * D_))[lane];
    float ss = v.x * v.x + v.y * v.y + v.z * v.z + v.w * v.w;
    #pragma unroll
    for (int off = 16; off; off >>= 1) ss += __shfl_xor(ss, off);
    if (lane == 0) {
        invn[row] = 1.0f / fmaxf(sqrtf(ss), EPS_NORM);
    }
}

// ---------------------------------------------------------------------------
// Kernel B: one workgroup (256 thr = 8 waves) per (b,s) pair.
//   - gather 32 normalized pull rows into LDS
//   - centroid k, membership bitmask
//   - pos: mean L2 of (pull - k + eps)
//   - neg: hinge(MARGIN - L1(fn_c - k + eps)) over non-members, via f32 WMMA
//     row-sum against a ones B-matrix.
// ---------------------------------------------------------------------------
__global__ __launch_bounds__(256)
void pp_main_kernel(const float* __restrict__ feats,
                    const int* __restrict__ sub_idx,
                    const float* __restrict__ invn,
                    float* __restrict__ pos_arr,
                    float* __restrict__ neg_arr) {
    __shared__ float    s_pull[L_][D_];   // 16 KB normalized pull rows
    __shared__ float    s_k[D_];          // centroid
    __shared__ unsigned s_mask[C_ / 32];  // membership bitset (512 bits)
    __shared__ int      s_idx[L_];
    __shared__ float    s_wpos[8];
    __shared__ float    s_wneg[8];

    const int bs   = blockIdx.x;          // 0 .. B*S-1
    const int b    = bs / S_;
    const int tid  = threadIdx.x;
    const int lane = tid & 31;
    const int wave = tid >> 5;

    if (tid < C_ / 32) s_mask[tid] = 0u;
    if (tid < L_)      s_idx[tid]  = sub_idx[bs * L_ + tid];
    __syncthreads();

    if (tid < L_) {
        const int c = s_idx[tid];
        atomicOr(&s_mask[c >> 5], 1u << (c & 31));
    }

    // Gather normalized pull rows: 32 rows x 32 float4 = 1024 float4 loads.
    for (int f = tid; f < L_ * (D_ / 4); f += 256) {
        const int row = f >> 5;          // 0..31
        const int c4  = f & 31;          // 0..31
        const int src = b * C_ + s_idx[row];
        const float4 v  = ((const float4*)(feats + (size_t)src * D_))[c4];
        const float inv = invn[src];
        float4 w;
        w.x = v.x * inv; w.y = v.y * inv; w.z = v.z * inv; w.w = v.w * inv;
        ((float4*)&s_pull[row][0])[c4] = w;
    }
    __syncthreads();

    // Centroid k[d] = mean_l pull[l][d]
    for (int d = tid; d < D_; d += 256) {
        float acc = 0.f;
        #pragma unroll 8
        for (int l = 0; l < L_; ++l) acc += s_pull[l][d];
        s_k[d] = acc * (1.0f / L_);
    }
    __syncthreads();

    // ---- pos: each wave handles rows l = wave, wave+8, ... -----------------
    float pos_acc = 0.f;
    for (int l = wave; l < L_; l += 8) {
        const int d0 = lane * 4;
        float ss = 0.f;
        #pragma unroll
        for (int j = 0; j < 4; ++j) {
            const float df = s_pull[l][d0 + j] - s_k[d0 + j] + EPS_PD;
            ss += df * df;
        }
        #pragma unroll
        for (int off = 16; off; off >>= 1) ss += __shfl_xor(ss, off);
        if (lane == 0) pos_acc += sqrtf(ss);
    }
    if (lane == 0) s_wpos[wave] = pos_acc;

    // ---- neg: L1 distances of all C rows to k, hinge at MARGIN -------------
    float neg_acc = 0.f;
    const int m     = lane & 15;   // A-matrix row within tile
    const int khalf = lane >> 4;   // 0: K=0,1  1: K=2,3  (f32 16x4 A layout)

#if USE_WMMA
    for (int tile = wave; tile < C_ / 16; tile += 8) {
        const int c0 = tile * 16;
        const float* rowp = feats + (size_t)(b * C_ + c0 + m) * D_;
        const float  rinv = invn[b * C_ + c0 + m];
        v8f acc = {};
        v2f ones; ones.x = 1.0f; ones.y = 1.0f;
        for (int d0 = 0; d0 < D_; d0 += 4) {
            const int d = d0 + 2 * khalf;
            v2f a;
            a.x = fabsf(rowp[d + 0] * rinv - s_k[d + 0] + EPS_PD);
            a.y = fabsf(rowp[d + 1] * rinv - s_k[d + 1] + EPS_PD);
            // D = A(16x4 absdiff) * B(4x16 ones) + C  -> running row sums
            acc = __builtin_amdgcn_wmma_f32_16x16x4_f32(
                false, a, false, ones, (short)0, acc, false, false);
        }
        // Row sums replicated across columns: lane 0 holds rows c0..c0+7 in
        // acc[0..7]; lane 16 holds rows c0+8..c0+15.
        if (m == 0) {
            #pragma unroll
            for (int e = 0; e < 8; ++e) {
                const int cc = c0 + e + 8 * khalf;
                const bool memb = (s_mask[cc >> 5] >> (cc & 31)) & 1u;
                if (!memb) {
                    const float h = MARGIN - acc[e];
                    if (h > 0.f) neg_acc += h;
                }
            }
        }
    }
    // fold lane16 partial into lane0
    neg_acc += __shfl_xor(neg_acc, 16);
#else
    for (int tile = wave; tile < C_ / 16; tile += 8) {
        const int c0 = tile * 16;
        for (int r = 0; r < 16; ++r) {
            const int cc = c0 + r;
            const float* rp  = feats + (size_t)(b * C_ + cc) * D_;
            const float  ri  = invn[b * C_ + cc];
            const int    d0  = lane * 4;
            float s = 0.f;
            #pragma unroll
            for (int j = 0; j < 4; ++j)
                s += fabsf(rp[d0 + j] * ri - s_k[d0 + j] + EPS_PD);
            #pragma unroll
            for (int off = 16; off; off >>= 1) s += __shfl_xor(s, off);
            if (lane == 0) {
                const bool memb = (s_mask[cc >> 5] >> (cc & 31)) & 1u;
                if (!memb) {
                    const float h = MARGIN - s;
                    if (h > 0.f) neg_acc += h;
                }
            }
        }
    }
#endif
    if (lane == 0) s_wneg[wave] = neg_acc;
    __syncthreads();

    if (tid == 0) {
        float p = 0.f, n = 0.f;
        #pragma unroll
        for (int w = 0; w < 8; ++w) { p += s_wpos[w]; n += s_wneg[w]; }
        int memb_cnt = 0;
        #pragma unroll
        for (int i = 0; i < C_ / 32; ++i) memb_cnt += __popc(s_mask[i]);
        const float push = (float)(C_ - memb_cnt);
        pos_arr[bs] = p * (1.0f / L_);
        neg_arr[bs] = n / push;
    }
}

// ---------------------------------------------------------------------------
// Kernel C: deterministic final mean over the B*S partials -> d_out[0..1]
// ---------------------------------------------------------------------------
__global__ __launch_bounds__(256)
void pp_finalize_kernel(const float* __restrict__ pos_arr,
                        const float* __restrict__ neg_arr,
                        float* __restrict__ out) {
    __shared__ float sp[256];
    __shared__ float sn[256];
    const int tid = threadIdx.x;
    float p = 0.f, n = 0.f;
    for (int i = tid; i < B_ * S_; i += 256) { p += pos_arr[i]; n += neg_arr[i]; }
    sp[tid] = p; sn[tid] = n;
    __syncthreads();
    for (int s = 128; s; s >>= 1) {
        if (tid < s) { sp[tid] += sp[tid + s]; sn[tid] += sn[tid + s]; }
        __syncthreads();
    }
    if (tid == 0) {
        out[0] = sp[0] * (1.0f / (B_ * S_));
        out[1] = sn[0] * (1.0f / (B_ * S_));
    }
}

// ---------------------------------------------------------------------------
extern "C" void kernel_launch(void* const* d_in, const int* in_sizes, int n_in,
                              void* d_out, int out_size, void* d_ws, size_t ws_size,
                              hipStream_t stream) {
    const float* feats   = (const float*)d_in[0];   // [B*C, D] f32
    const int*   sub_idx = (const int*)d_in[1];     // [B, S, L] i32
    (void)in_sizes; (void)n_in; (void)out_size; (void)ws_size;

    float* invn    = (float*)d_ws;                  // B*C floats (256 KB)
    float* pos_arr = invn + (size_t)B_ * C_;        // B*S floats
    float* neg_arr = pos_arr + B_ * S_;             // B*S floats

    pp_rownorm_kernel<<<(B_ * C_) / 8, 256, 0, stream>>>(feats, invn, B_ * C_);
    pp_main_kernel<<<B_ * S_, 256, 0, stream>>>(feats, sub_idx, invn,
                                                pos_arr, neg_arr);
    pp_finalize_kernel<<<1, 256, 0, stream>>>(pos_arr, neg_arr, (float*)d_out);
}